// ScaleTransfer_10118942949508
// MI455X (gfx1250) — compile-verified
//
#include <hip/hip_runtime.h>
#include <cstdint>

// Depth-to-space (x-major sub-pixel): out[b,c,4*yb+yr,4*xb+xr] = in[b, c*16+4*xr+yr, yb, xb]
// in : (16, 4096, 32, 32) f32   out : (16, 256, 128, 128) f32
// Pure HBM-streaming kernel: 512 MB total traffic -> ~22 us floor @ 23.3 TB/s.
// CDNA5 path: async global->LDS b128 loads (ASYNCcnt), LDS lane-redistribution,
// NT b128 stores, double-buffered on in-order async-load completion.

typedef float v4f __attribute__((ext_vector_type(4)));

#define R_PB        8                    // rows per batch per wave
#define WAVES_PB    8                    // 256 threads = 8 wave32
#define NBLOCKS     2048
#define OUT_W       128
#define TOTAL_ROWS  (16 * 256 * 128)     // 524288 output rows of 128 floats

__device__ __forceinline__ void issue_batch_loads(const float* __restrict__ in,
                                                  int t, unsigned lds_base, int lane) {
    // Row r: y = r & 127, bc = r >> 7 (bc = b*256+c), yr = y&3, yb = y>>2.
    // Lane l loads 16B from channel chunk (l>>3): element offset
    //   (bc*16 + yr)*1024 + (l>>3)*4096 + yb*32 + (l&7)*4
    // into LDS at lds_base + j*512 + l*16  (channel-major block: dword d = xr*32 + xb).
#pragma unroll
    for (int j = 0; j < R_PB; ++j) {
        const int row = t * R_PB + j;
        const int y   = row & (OUT_W - 1);
        const int bc  = row >> 7;
        const int yr  = y & 3;
        const int yb  = y >> 2;
        const float* g = in
            + (size_t)(bc * 16 + yr) * 1024
            + (size_t)((lane >> 3) * 4096)
            + (size_t)(yb * 32)
            + (size_t)((lane & 7) * 4);
        const unsigned l = lds_base + (unsigned)(j * 512 + lane * 16);
        asm volatile("global_load_async_to_lds_b128 %0, %1, off"
                     :: "v"(l), "v"(g)
                     : "memory");
    }
}

__device__ __forceinline__ void store_batch(float* __restrict__ out,
                                            const float* __restrict__ buf,
                                            int t, int lane) {
#pragma unroll
    for (int j = 0; j < R_PB; ++j) {
        const int row   = t * R_PB + j;
        const float* sj = buf + j * OUT_W;
        v4f v;                          // output x = 4*lane + xr  <-  LDS dword xr*32 + lane
        v.x = sj[      lane];
        v.y = sj[32 +  lane];
        v.z = sj[64 +  lane];
        v.w = sj[96 +  lane];
        v4f* dst = reinterpret_cast<v4f*>(out + (size_t)row * OUT_W) + lane;
        __builtin_nontemporal_store(v, dst);
    }
}

__global__ __launch_bounds__(256) void d2s_async(const float* __restrict__ in,
                                                 float* __restrict__ out) {
    // 8 waves * 2 buffers * 8 rows * 512B = 64 KB LDS (of 320 KB/WGP)
    __shared__ float smem[WAVES_PB * 2 * R_PB * OUT_W];

    const int lane   = threadIdx.x & 31;
    const int wid    = threadIdx.x >> 5;
    const int gw     = blockIdx.x * WAVES_PB + wid;
    const int nwaves = gridDim.x * WAVES_PB;
    const int nb     = TOTAL_ROWS / R_PB;          // 65536 batches

    float* wbase = &smem[wid * (2 * R_PB * OUT_W)];
    const unsigned lds0 = (unsigned)reinterpret_cast<uintptr_t>(wbase);
    const unsigned lds1 = lds0 + (unsigned)(R_PB * OUT_W * 4);

    int t = gw;
    if (t >= nb) return;

    int buf = 0;
    issue_batch_loads(in, t, lds0, lane);
    for (;;) {
        const int  tn   = t + nwaves;
        const bool more = (tn < nb);
        if (more) {
            issue_batch_loads(in, tn, buf ? lds0 : lds1, lane);
            // ASYNCcnt tracks only our async loads, which complete in order:
            // <= R_PB outstanding  =>  previous batch fully landed in LDS.
            asm volatile("s_wait_asynccnt %0" :: "i"(R_PB) : "memory");
        } else {
            asm volatile("s_wait_asynccnt 0" ::: "memory");
        }
        store_batch(out, wbase + buf * (R_PB * OUT_W), t, lane);
        if (!more) break;
        t = tn;
        buf ^= 1;
    }
}

extern "C" void kernel_launch(void* const* d_in, const int* in_sizes, int n_in,
                              void* d_out, int out_size, void* d_ws, size_t ws_size,
                              hipStream_t stream) {
    (void)in_sizes; (void)n_in; (void)out_size; (void)d_ws; (void)ws_size;
    const float* in = reinterpret_cast<const float*>(d_in[0]);
    float* out      = reinterpret_cast<float*>(d_out);
    d2s_async<<<dim3(NBLOCKS), dim3(256), 0, stream>>>(in, out);
}